// LightRegularizedGAT_10514079941120
// MI455X (gfx1250) — compile-verified
//
#include <hip/hip_runtime.h>
#include <hip/hip_bf16.h>
#include <math.h>

#define BATCH 8
#define NNODE 1024
#define FIN   256
#define FOUT  256
#define HEADS 4
#define DHEAD 64

typedef float v2f __attribute__((ext_vector_type(2)));
typedef float v8f __attribute__((ext_vector_type(8)));

// ---------------------------------------------------------------------------
// CDNA5 async Global->LDS DMA (ASYNCcnt-tracked), GVS mode:
//   copies 16 bytes per lane from global address base plus g_off to LDS offset lds_off
// Inline asm per CDNA5_HIP.md guidance (builtin arity differs per toolchain).
// ---------------------------------------------------------------------------
__device__ __forceinline__ void async_g2l_b128(unsigned lds_off, unsigned g_off,
                                               const void* base)
{
    asm volatile("global_load_async_to_lds_b128 %0, %1, %2"
                 :: "v"(lds_off), "v"(g_off), "s"(base)
                 : "memory");
}
__device__ __forceinline__ void wait_async0()
{
    asm volatile("s_wait_asynccnt 0" ::: "memory");
}

// ---------------------------------------------------------------------------
// Kernel A: h = x @ W  (WMMA f32 16x16x4), fused LayerNorm, e_i / e_j dots.
// grid = B * (N/16) = 512 blocks, 512 threads (16 waves).
// Wave w computes the 16x16 output tile at columns [16w, 16w+16).
// ---------------------------------------------------------------------------
__global__ __launch_bounds__(512)
void gat_gemm_ln_kernel(const float* __restrict__ x,
                        const float* __restrict__ W,
                        const float* __restrict__ gamma,
                        const float* __restrict__ beta,
                        const float* __restrict__ a,
                        float* __restrict__ hn,   // [B][N][FOUT] normalized h
                        float* __restrict__ ei,   // [B][N][H]
                        float* __restrict__ ej)   // [B][N][H]
{
    __shared__ float xs[16][FIN + 1];   // padded: bank-conflict-free column reads
    __shared__ float hs[16][FOUT + 1];

    const int blk  = blockIdx.x;
    const int b    = blk >> 6;           // 64 row-tiles per batch image
    const int i0   = (blk & 63) << 4;
    const int t    = threadIdx.x;
    const int lane = t & 31;
    const int w    = t >> 5;

    // ---- async-DMA 16 x 256 block of x into LDS (b128, ASYNCcnt) ----
    const float* xblk = x + ((size_t)b * NNODE + i0) * FIN;
    {
        const unsigned lbase = (unsigned)(uintptr_t)&xs[0][0];
        for (int idx = t; idx < (16 * FIN) / 4; idx += 512) {   // 1024 chunks
            int r = idx >> 6;                 // 64 x b128 per row
            int c = (idx & 63) << 2;
            async_g2l_b128(lbase + (unsigned)((r * (FIN + 1) + c) * 4),
                           (unsigned)((r * FIN + c) * 4), xblk);
        }
    }
    wait_async0();
    __syncthreads();

    // ---- WMMA GEMM: acc(16x16) += A(16x4) * B(4x16), K = 256 ----
    const int col0  = w << 4;
    const int mrow  = lane & 15;            // A: M row / B: N col
    const int khalf = (lane >> 4) << 1;     // 0 or 2 (K pair select)
    v8f acc = {0.f, 0.f, 0.f, 0.f, 0.f, 0.f, 0.f, 0.f};
    for (int k = 0; k < FIN; k += 4) {
        v2f af, bf;
        af[0] = xs[mrow][k + khalf + 0];
        af[1] = xs[mrow][k + khalf + 1];
        const float* wp = W + (size_t)(k + khalf) * FOUT + col0 + mrow;
        bf[0] = wp[0];
        bf[1] = wp[FOUT];
        acc = __builtin_amdgcn_wmma_f32_16x16x4_f32(
            false, af, false, bf, (short)0, acc, false, false);
    }

    // ---- spill C tile to LDS (row = v + 8*(lane/16), col = col0 + lane%16) ----
#pragma unroll
    for (int v = 0; v < 8; ++v) {
        int row = v + ((lane >> 4) << 3);
        hs[row][col0 + mrow] = acc[v];
    }
    __syncthreads();

    // ---- LayerNorm: wave w owns row w ----
    {
        const int r = w;
        float s = 0.f, sq = 0.f;
        for (int c = lane; c < FOUT; c += 32) {
            float v = hs[r][c]; s += v; sq += v * v;
        }
#pragma unroll
        for (int off = 16; off; off >>= 1) {
            s  += __shfl_xor(s,  off, 32);
            sq += __shfl_xor(sq, off, 32);
        }
        const float mu  = s * (1.f / FOUT);
        const float var = sq * (1.f / FOUT) - mu * mu;
        const float rs  = rsqrtf(var + 1e-5f);
        float* hrow = hn + ((size_t)b * NNODE + i0 + r) * FOUT;
        for (int c = lane; c < FOUT; c += 32) {
            float v = (hs[r][c] - mu) * rs * gamma[c] + beta[c];
            hs[r][c] = v;
            hrow[c]  = v;
        }
    }
    __syncthreads();

    // ---- attention logit dots: 16 rows x 4 heads x {src,dst} = 128 tasks ----
    if (t < 128) {
        const int r     = t >> 3;
        const int hd    = (t >> 1) & 3;
        const int which = t & 1;             // 0 -> a_src, 1 -> a_dst
        const float* av = a + which * DHEAD;
        float s = 0.f;
        for (int d = 0; d < DHEAD; ++d)
            s += hs[r][hd * DHEAD + d] * av[d];
        float* dst = which ? ej : ei;
        dst[(((size_t)b * NNODE + i0 + r) << 2) + hd] = s;
    }
}

// ---------------------------------------------------------------------------
// Kernel B: masked leaky-ReLU attention + softmax + alpha @ h  (WMMA f32).
// grid = B * (N/16) = 512 blocks, 512 threads (16 waves).
// ej table for the whole batch row is async-DMA'd to LDS once per block.
// Pass 1: wave w scans row i0+w over all j -> (max, sum) per head.
// Pass 2: wave w = (head, dtile) accumulates its 16x16 (i x d) output tile.
// ---------------------------------------------------------------------------
__global__ __launch_bounds__(512)
void gat_attn_kernel(const float* __restrict__ adj,
                     const float* __restrict__ hn,
                     const float* __restrict__ ei,
                     const float* __restrict__ ej,
                     float* __restrict__ out)
{
    __shared__ float ejs[NNODE][4];  // 16 KB: e_j for every j, all heads
    __shared__ float sm[16][4];      // row-wise max
    __shared__ float sinv[16][4];    // 1 / row-wise sum
    __shared__ float eis[16][4];     // e_i for the 16 rows

    const int blk  = blockIdx.x;
    const int b    = blk >> 6;
    const int i0   = (blk & 63) << 4;
    const int t    = threadIdx.x;
    const int lane = t & 31;
    const int w    = t >> 5;

    const float* adjb = adj + (size_t)b * NNODE * NNODE;
    const float* ejb  = ej + ((size_t)b * NNODE << 2);
    const float* hnb  = hn + (size_t)b * NNODE * FOUT;

    // ---- async-DMA ej[b][*][*] (16 KB) into LDS (b128 per node) ----
    {
        const unsigned lbase = (unsigned)(uintptr_t)&ejs[0][0];
        for (int idx = t; idx < NNODE; idx += 512) {
            unsigned off = (unsigned)idx << 4;       // 16 B per node
            async_g2l_b128(lbase + off, off, ejb);
        }
    }
    wait_async0();
    __syncthreads();

    // ---------------- pass 1: online (max, sum) per (row, head) --------------
    {
        const int r = w;
        const float* eip = ei + (((size_t)b * NNODE + i0 + r) << 2);
        float eiv[4];
#pragma unroll
        for (int hd = 0; hd < 4; ++hd) eiv[hd] = eip[hd];

        float m[4], s[4];
#pragma unroll
        for (int hd = 0; hd < 4; ++hd) { m[hd] = -INFINITY; s[hd] = 0.f; }

        const float* arow = adjb + (size_t)(i0 + r) * NNODE;
        for (int j = lane; j < NNODE; j += 32) {
            const float adjv = arow[j];
#pragma unroll
            for (int hd = 0; hd < 4; ++hd) {
                float e = eiv[hd] + ejs[j][hd];
                e = e > 0.f ? e : 0.2f * e;          // leaky ReLU
                e = (adjv == 0.f) ? -1e9f : e;       // adjacency mask
                float mn = fmaxf(m[hd], e);
                s[hd] = s[hd] * __expf(m[hd] - mn) + __expf(e - mn);
                m[hd] = mn;
            }
        }
#pragma unroll
        for (int off = 16; off; off >>= 1) {
#pragma unroll
            for (int hd = 0; hd < 4; ++hd) {
                float mo = __shfl_xor(m[hd], off, 32);
                float so = __shfl_xor(s[hd], off, 32);
                float mn = fmaxf(m[hd], mo);
                s[hd] = s[hd] * __expf(m[hd] - mn) + so * __expf(mo - mn);
                m[hd] = mn;
            }
        }
        if (lane == 0) {
#pragma unroll
            for (int hd = 0; hd < 4; ++hd) {
                sm[r][hd]   = m[hd];
                sinv[r][hd] = 1.f / s[hd];
                eis[r][hd]  = eiv[hd];
            }
        }
    }
    __syncthreads();

    // ---------------- pass 2: acc(i,d) += P(i,j) @ V(j,d) via WMMA ------------
    const int hd    = w >> 2;               // head 0..3
    const int dt    = w & 3;                // 16-wide d tile 0..3
    const int mrow  = lane & 15;            // A: i row / B: n col
    const int khalf = (lane >> 4) << 1;     // K pair select within K=4 step

    const float mi  = sm[mrow][hd];
    const float li  = sinv[mrow][hd];
    const float eii = eis[mrow][hd];
    const float* arow = adjb + (size_t)(i0 + mrow) * NNODE;
    const float* vcol = hnb + hd * DHEAD + (dt << 4) + mrow;

    v8f acc = {0.f, 0.f, 0.f, 0.f, 0.f, 0.f, 0.f, 0.f};
    for (int jb = 0; jb < NNODE / 16; ++jb) {
#pragma unroll
        for (int kk = 0; kk < 4; ++kk) {
            const int jbase = (jb << 4) + (kk << 2) + khalf;
            v2f af, bf;
#pragma unroll
            for (int v = 0; v < 2; ++v) {
                const int j = jbase + v;
                // P(i, j): masked leaky logit -> normalized softmax weight
                float e = eii + ejs[j][hd];
                e = e > 0.f ? e : 0.2f * e;
                e = (arow[j] == 0.f) ? -1e9f : e;    // exp(-1e9 - m) == 0
                af[v] = __expf(e - mi) * li;
                // V(j, n): normalized h, column slice for this (head, dtile)
                bf[v] = vcol[(size_t)j * FOUT];
            }
            acc = __builtin_amdgcn_wmma_f32_16x16x4_f32(
                false, af, false, bf, (short)0, acc, false, false);
        }
    }

    // ---- store 16x16 output tile ----
    float* ob = out + ((size_t)b * NNODE + i0) * FOUT + hd * DHEAD + (dt << 4) + mrow;
#pragma unroll
    for (int v = 0; v < 8; ++v) {
        int row = v + ((lane >> 4) << 3);
        ob[(size_t)row * FOUT] = acc[v];
    }
}

// ---------------------------------------------------------------------------
extern "C" void kernel_launch(void* const* d_in, const int* in_sizes, int n_in,
                              void* d_out, int out_size, void* d_ws, size_t ws_size,
                              hipStream_t stream)
{
    const float* x     = (const float*)d_in[0];
    const float* adj   = (const float*)d_in[1];
    const float* W     = (const float*)d_in[2];
    const float* gamma = (const float*)d_in[3];
    const float* beta  = (const float*)d_in[4];
    const float* a     = (const float*)d_in[5];
    float* out = (float*)d_out;

    char* ws = (char*)d_ws;
    float* hn = (float*)ws;                                       // 8 MB
    float* ei = (float*)(ws + (size_t)BATCH * NNODE * FOUT * 4);  // 128 KB
    float* ej = ei + (size_t)BATCH * NNODE * HEADS;               // 128 KB

    dim3 grid(BATCH * (NNODE / 16));   // 512 blocks
    dim3 block(512);                   // 16 wave32 per block

    gat_gemm_ln_kernel<<<grid, block, 0, stream>>>(x, W, gamma, beta, a, hn, ei, ej);
    gat_attn_kernel<<<grid, block, 0, stream>>>(adj, hn, ei, ej, out);
}